// ProgressiveSelector_76982993814147
// MI455X (gfx1250) — compile-verified
//
#include <hip/hip_runtime.h>
#include <hip/hip_bf16.h>
#include <stdint.h>

// ---------------------------------------------------------------------------
// ProgressiveSelector: per-row causal top-k mask.
//   B=8, S=2048, current_k = min(S, 16 + (240*min(step,1000))/1000)
//   out[0 .. B*S*S)        : mask as float (1.0 selected / 0.0 not)
//   out[B*S*S .. +B*S)     : k_values as float (current_k broadcast)
// Memory-bound (~268 MB @ 23.3 TB/s ~= 11.5 us floor). Selection runs out of
// LDS after a single async global->LDS copy per row (CDNA5 ASYNCcnt path),
// radix-select with LDS histograms + wave-level (wave32 shfl) suffix scans.
// ---------------------------------------------------------------------------

#define B_DIM     8
#define S_DIM     2048
#define START_K   16
#define END_K     256
#define MAX_STEPS 1000
#define THREADS   256
#define PER_THR   (S_DIM / THREADS)   // 8 elements per thread

// Async global->LDS b128 builtin wants v4i pointers in AS1 (global) / AS3 (LDS).
typedef int v4i __attribute__((vector_size(16)));
typedef __attribute__((address_space(1))) v4i g_v4i;
typedef __attribute__((address_space(3))) v4i l_v4i;

// Order-preserving float -> uint key (larger float => larger key).
// Key 0 is reserved for "invalid" (only -NaN(all-ones) could collide).
__device__ __forceinline__ unsigned int f2key(float f) {
    unsigned int u = __float_as_uint(f);
    return (u & 0x80000000u) ? ~u : (u | 0x80000000u);
}

__global__ __launch_bounds__(THREADS)
void ProgressiveSelector_76982993814147_kernel(
        const float* __restrict__ scores,   // [B, S, S]
        const int*   __restrict__ step_ptr, // [1]
        float*       __restrict__ out)      // [B*S*S + B*S]
{
    __shared__ float        srow[S_DIM];
    __shared__ unsigned int hist[256];
    __shared__ unsigned int scan[257];     // suffix sums over bins
    __shared__ unsigned int wtot[8];       // per-wave (32-bin chunk) totals
    __shared__ unsigned int sel_bin;
    __shared__ unsigned int sel_above;
    __shared__ unsigned int tie_ctr;

    const int row  = blockIdx.x;           // 0 .. B*S-1
    const int q    = row & (S_DIM - 1);    // query index within sequence
    const int tid  = threadIdx.x;
    const int lane = tid & 31;             // wave32
    const int wid  = tid >> 5;             // 8 waves per block

    // Progressive-k schedule — integer-exact (truncation == floor for >=0).
    const int step = step_ptr[0];
    int sc = step < MAX_STEPS ? step : MAX_STEPS;
    int ck = START_K + ((END_K - START_K) * sc) / MAX_STEPS;
    ck     = ck < S_DIM ? ck : S_DIM;
    const int n = q + 1;                   // number of valid (causal) keys

    float* orow = out + (size_t)row * S_DIM;

    if (tid == 0) {
        out[(size_t)B_DIM * S_DIM * S_DIM + row] = (float)ck;
    }

    // ---- Easy case: everything causal is selected (q+1 <= k) --------------
    if (n <= ck) {
        #pragma unroll
        for (int i = 0; i < PER_THR; ++i) {
            int j = tid + i * THREADS;
            __builtin_nontemporal_store((j <= q) ? 1.0f : 0.0f, &orow[j]);
        }
        return;  // uniform branch: whole block exits together
    }

    // ---- Stage the row into LDS via async global->LDS DMA (ASYNCcnt) ------
    {
        const float* g = scores + (size_t)row * S_DIM + tid * PER_THR;
        float*       l = srow + tid * PER_THR;
#if __has_builtin(__builtin_amdgcn_global_load_async_to_lds_b128)
        __builtin_amdgcn_global_load_async_to_lds_b128(
            (g_v4i*)g,       (l_v4i*)l,       0, 0);
        __builtin_amdgcn_global_load_async_to_lds_b128(
            (g_v4i*)(g + 4), (l_v4i*)(l + 4), 0, 0);
#if __has_builtin(__builtin_amdgcn_s_wait_asynccnt)
        __builtin_amdgcn_s_wait_asynccnt(0);
#else
        asm volatile("s_wait_asynccnt 0" ::: "memory");
#endif
#else
        #pragma unroll
        for (int i = 0; i < PER_THR; ++i) l[i] = __builtin_nontemporal_load(&g[i]);
#endif
    }
    __syncthreads();

    // ---- Cache keys in VGPRs (strided so final stores coalesce) -----------
    unsigned int key[PER_THR];
    #pragma unroll
    for (int i = 0; i < PER_THR; ++i) {
        int j = tid + i * THREADS;
        key[i] = (j < n) ? f2key(srow[j]) : 0u;
    }

    // ---- Radix-select the ck-th largest key (4 x 8-bit passes) -------------
    unsigned int prefix    = 0u;
    unsigned int remaining = (unsigned int)ck;   // here n > ck

    #pragma unroll
    for (int p = 3; p >= 0; --p) {
        const unsigned int hi_mask =
            (p == 3) ? 0u : (0xFFFFFFFFu << (8 * (p + 1)));

        hist[tid] = 0u;
        __syncthreads();

        #pragma unroll
        for (int i = 0; i < PER_THR; ++i) {
            unsigned int k = key[i];
            if (k != 0u && (k & hi_mask) == (prefix & hi_mask)) {
                atomicAdd(&hist[(k >> (8 * p)) & 0xFFu], 1u);  // ds_add_u32
            }
        }
        __syncthreads();

        // Suffix sums over the 256 bins, hierarchically:
        //   (a) in-wave inclusive suffix scan of each 32-bin chunk (shfl, no
        //       barriers — wave32 is execution-synchronous),
        //   (b) chunk totals through LDS, each thread adds suffix of the
        //       chunks above its own.
        unsigned int c = hist[tid];
        #pragma unroll
        for (int d = 1; d < 32; d <<= 1) {
            unsigned int t = __shfl_down(c, d, 32);
            if (lane + d < 32) c += t;
        }
        if (lane == 0) wtot[wid] = c;      // total of this 32-bin chunk
        __syncthreads();

        unsigned int above = 0u;
        for (int w = wid + 1; w < 8; ++w) above += wtot[w];
        // note: lane l's c already holds suffix within chunk starting at bin tid
        scan[tid] = ((lane == 0) ? c : c) + above;   // suffix_sum(tid)
        if (tid == 0) scan[256] = 0u;
        __syncthreads();

        // Largest bin b with suffix_sum(b) >= remaining (scan non-increasing).
        unsigned int my  = scan[tid];
        unsigned int nxt = scan[tid + 1];
        if (my >= remaining && nxt < remaining) {
            sel_bin   = (unsigned int)tid;
            sel_above = nxt;               // count of keys strictly above bin b
        }
        __syncthreads();

        prefix    |= (sel_bin << (8 * p));
        remaining -= sel_above;
        __syncthreads();
    }
    // prefix == exact key value of the ck-th largest; `remaining` keys equal
    // to it still need to be taken.

    // ---- Emit mask row (streaming NT stores, ties via LDS counter) ---------
    if (tid == 0) tie_ctr = 0u;
    __syncthreads();

    #pragma unroll
    for (int i = 0; i < PER_THR; ++i) {
        int j = tid + i * THREADS;
        unsigned int k = key[i];
        float v = 0.0f;
        if (k > prefix) {
            v = 1.0f;
        } else if (k == prefix && k != 0u) {
            unsigned int old = atomicAdd(&tie_ctr, 1u);  // ds_add_rtn_u32
            if (old < remaining) v = 1.0f;
        }
        __builtin_nontemporal_store(v, &orow[j]);
    }
}

extern "C" void kernel_launch(void* const* d_in, const int* in_sizes, int n_in,
                              void* d_out, int out_size, void* d_ws, size_t ws_size,
                              hipStream_t stream) {
    // d_in[0]: x [B,S,D] (unused — only shapes matter)
    // d_in[1]: index_scores [B,S,S] float32
    // d_in[2]: training_step (single int)
    const float* scores = (const float*)d_in[1];
    const int*   step   = (const int*)d_in[2];
    float*       out    = (float*)d_out;

    dim3 grid(B_DIM * S_DIM);
    dim3 block(THREADS);
    ProgressiveSelector_76982993814147_kernel<<<grid, block, 0, stream>>>(
        scores, step, out);
}